// GConvLSTMRegression_29437705847599
// MI455X (gfx1250) — compile-verified
//
#include <hip/hip_runtime.h>
#include <hip/hip_bf16.h>

// Problem constants (from reference)
#define NN   100000
#define EE   1600000
#define FIN  128
#define HIDN 256

typedef __attribute__((ext_vector_type(2))) float v2f;
typedef __attribute__((ext_vector_type(8))) float v8f;

__device__ __forceinline__ float sigm(float x) { return 1.0f / (1.0f + __expf(-x)); }

// ---------------- graph preprocessing ----------------

__global__ void deg_kernel(const long long* __restrict__ ei, float* __restrict__ deg) {
    int e = blockIdx.x * blockDim.x + threadIdx.x;
    if (e < EE) atomicAdd(&deg[(int)ei[e]], 1.0f);
}

__global__ void dis_kernel(float* __restrict__ deg) {  // in-place deg -> d^{-1/2}
    int i = blockIdx.x * blockDim.x + threadIdx.x;
    if (i < NN) {
        float d = deg[i];
        deg[i] = (d > 0.0f) ? rsqrtf(d) : 0.0f;
    }
}

__global__ void norm_kernel(const long long* __restrict__ ei, const float* __restrict__ dis,
                            float* __restrict__ norm) {
    int e = blockIdx.x * blockDim.x + threadIdx.x;
    if (e < EE) {
        int s = (int)ei[e];
        int d = (int)ei[EE + e];
        norm[e] = -dis[s] * dis[d];
    }
}

// out[dst] += norm_e * in[src]  (one wave = one edge, lane = 4 features; fully coalesced)
__global__ void prop_kernel(const long long* __restrict__ ei, const float* __restrict__ norm,
                            const float* __restrict__ vin, float* __restrict__ vout) {
    unsigned gid = blockIdx.x * blockDim.x + threadIdx.x;
    if (gid >= (unsigned)EE * 32u) return;
    int e = (int)(gid >> 5);
    int c = (int)(gid & 31u);
    int s = (int)ei[e];
    int d = (int)ei[EE + e];
    float w = norm[e];
    float4 v = ((const float4*)vin)[(size_t)s * 32 + c];
    float* o = vout + (size_t)d * FIN + c * 4;
    atomicAdd(o + 0, w * v.x);
    atomicAdd(o + 1, w * v.y);
    atomicAdd(o + 2, w * v.z);
    atomicAdd(o + 3, w * v.w);
}

__global__ void cheb2_finalize(const float* __restrict__ x, float* __restrict__ Tx2) {
    int i = blockIdx.x * blockDim.x + threadIdx.x;
    if (i < NN * FIN) Tx2[i] = 2.0f * Tx2[i] - x[i];
}

// ---------------- weight pre-pack into WMMA fragment order ----------------
// Pk2[(((ks*3 + gg)*16 + ht)*32 + lane)] = { W[g][cheb][f+2*half][col],
//                                           W[g][cheb][f+2*half+1][col] }
// so every B-fragment is a single coalesced 256B float2 (b64) load.
__global__ void pack_weights(const float* __restrict__ Wx, float2* __restrict__ Pk2) {
    int o2 = blockIdx.x * blockDim.x + threadIdx.x;
    if (o2 >= 96 * 3 * 16 * 32) return;
    int lane = o2 & 31;
    int ht   = (o2 >> 5) & 15;
    int gg   = (o2 >> 9) % 3;
    int ks   = o2 / (512 * 3);
    int k0   = ks * 4;
    int half = lane >> 4;
    int cc   = lane & 15;
    int cheb = k0 >> 7;
    int f    = k0 & 127;
    int col  = ht * 16 + cc;
    int g    = (gg == 0) ? 0 : (gg == 1 ? 2 : 3);  // live gates i, c, o
    const float* wb = Wx + ((size_t)(g * 3 + cheb) * FIN + f + 2 * half) * HIDN;
    Pk2[o2] = make_float2(wb[col], wb[HIDN + col]);
}

// ---------------- fused GEMM + LSTM gates + output projection ----------------
// One block = 32 nodes (two 16-row tiles), 8 waves. Wave w owns hidden units
// [w*32, w*32+32) across the three live gates {i, c, o}. A-tile [Tx0|Tx1|Tx2]
// (32 x 384 fp32) staged in LDS via GLOBAL_LOAD_ASYNC_TO_LDS_B128 (ASYNCcnt).
// Inner loop: 96 K-steps, 12 v_wmma_f32_16x16x4_f32 per step per wave, with
// B-fragments double-buffered (loads for ks+1 issued before WMMAs of ks).

#define AP 388  // padded LDS row stride in floats; 388*4 B = 97*16 B (b128-aligned, bank-safe)

__device__ __forceinline__ void loadB(const float2* __restrict__ Pk2,
                                      int ks, int wave, int lane, v2f b[6]) {
#pragma unroll
    for (int gg = 0; gg < 3; ++gg) {
#pragma unroll
        for (int t = 0; t < 2; ++t) {
            float2 bb = Pk2[((ks * 3 + gg) * 16 + wave * 2 + t) * 32 + lane];
            b[gg * 2 + t].x = bb.x;
            b[gg * 2 + t].y = bb.y;
        }
    }
}

__launch_bounds__(256)
__global__ void fused_gemm_gates(const float* __restrict__ x,
                                 const float* __restrict__ Tx1,
                                 const float* __restrict__ Tx2,
                                 const float2* __restrict__ Pk2,
                                 const float* __restrict__ bx,
                                 const float* __restrict__ bh,
                                 const float* __restrict__ w_c,
                                 const float* __restrict__ b_gate,
                                 const float* __restrict__ lin_w,
                                 const float* __restrict__ lin_b,
                                 float* __restrict__ out) {
    __shared__ float At[32 * AP];
    __shared__ float part[32];

    const int tid   = threadIdx.x;
    const int node0 = blockIdx.x * 32;

    // Async-stage the A tile: 32 rows x 96 float4 chunks = 3072, 12 per thread.
#pragma unroll
    for (int i = 0; i < 12; ++i) {
        int idx = tid + i * 256;
        int row = idx / 96;
        int q   = idx % 96;            // float4 index within row
        int seg = q >> 5;              // 0: x, 1: Tx1, 2: Tx2
        int cc  = q & 31;
        const float* src = (seg == 0) ? x : (seg == 1 ? Tx1 : Tx2);
        unsigned long long gaddr =
            (unsigned long long)(uintptr_t)(src + ((size_t)(node0 + row) * 32 + cc) * 4);
        unsigned ldsoff = (unsigned)(uintptr_t)&At[row * AP + q * 4];
        asm volatile("global_load_async_to_lds_b128 %0, %1, off"
                     :: "v"(ldsoff), "v"(gaddr) : "memory");
    }
    if (tid < 32) part[tid] = 0.0f;
    asm volatile("s_wait_asynccnt 0x0" ::: "memory");
    __syncthreads();

    const int wave = tid >> 5;
    const int lane = tid & 31;
    const int half = lane >> 4;   // K-half selector for A/B fragments
    const int c    = lane & 15;   // row (A) / column (B,C,D) within tile
    const int h0   = wave * 32;   // this wave's hidden-unit base

    v8f acc[2][3][2];             // [row-tile][gate][hid-tile]
#pragma unroll
    for (int rt = 0; rt < 2; ++rt)
#pragma unroll
        for (int gg = 0; gg < 3; ++gg)
#pragma unroll
            for (int t = 0; t < 2; ++t)
#pragma unroll
                for (int r = 0; r < 8; ++r) acc[rt][gg][t][r] = 0.0f;

    v2f bcur[6], bnext[6];
    loadB(Pk2, 0, wave, lane, bcur);

    for (int ks = 0; ks < 95; ++ks) {
        const int k0 = ks * 4;
        v2f a0, a1;  // A fragments (16x4): lane holds M=c, K = k0 + 2*half + {0,1}
        a0.x = At[c * AP + k0 + 2 * half + 0];
        a0.y = At[c * AP + k0 + 2 * half + 1];
        a1.x = At[(16 + c) * AP + k0 + 2 * half + 0];
        a1.y = At[(16 + c) * AP + k0 + 2 * half + 1];

        loadB(Pk2, ks + 1, wave, lane, bnext);  // prefetch next step's 6 fragments

#pragma unroll
        for (int gg = 0; gg < 3; ++gg) {
#pragma unroll
            for (int t = 0; t < 2; ++t) {
                acc[0][gg][t] = __builtin_amdgcn_wmma_f32_16x16x4_f32(
                    false, a0, false, bcur[gg * 2 + t], (short)0, acc[0][gg][t], false, false);
                acc[1][gg][t] = __builtin_amdgcn_wmma_f32_16x16x4_f32(
                    false, a1, false, bcur[gg * 2 + t], (short)0, acc[1][gg][t], false, false);
            }
        }
#pragma unroll
        for (int j = 0; j < 6; ++j) bcur[j] = bnext[j];
    }
    {   // epilogue: ks = 95
        const int k0 = 95 * 4;
        v2f a0, a1;
        a0.x = At[c * AP + k0 + 2 * half + 0];
        a0.y = At[c * AP + k0 + 2 * half + 1];
        a1.x = At[(16 + c) * AP + k0 + 2 * half + 0];
        a1.y = At[(16 + c) * AP + k0 + 2 * half + 1];
#pragma unroll
        for (int gg = 0; gg < 3; ++gg) {
#pragma unroll
            for (int t = 0; t < 2; ++t) {
                acc[0][gg][t] = __builtin_amdgcn_wmma_f32_16x16x4_f32(
                    false, a0, false, bcur[gg * 2 + t], (short)0, acc[0][gg][t], false, false);
                acc[1][gg][t] = __builtin_amdgcn_wmma_f32_16x16x4_f32(
                    false, a1, false, bcur[gg * 2 + t], (short)0, acc[1][gg][t], false, false);
            }
        }
    }

    // Gates (H = C = 0 initial state => gh = bh, F*C = 0):
    //   I = sigmoid(acc_i + bx0 + bh0 + bg0);  T = tanh(acc_c + bx2 + bh2 + bg2)
    //   C = I*T;  O = sigmoid(acc_o + bx3 + bh3 + w_c[2]*C + bg3)
    //   H = O*tanh(C);  partial[node] += H*lin_w[col]
    float psum[2][8];
#pragma unroll
    for (int rt = 0; rt < 2; ++rt)
#pragma unroll
        for (int r = 0; r < 8; ++r) psum[rt][r] = 0.0f;

#pragma unroll
    for (int t = 0; t < 2; ++t) {
        const int col = h0 + t * 16 + c;
        const float bi  = bx[0 * HIDN + col] + bh[0 * HIDN + col] + b_gate[0 * HIDN + col];
        const float bc  = bx[2 * HIDN + col] + bh[2 * HIDN + col] + b_gate[2 * HIDN + col];
        const float bo  = bx[3 * HIDN + col] + bh[3 * HIDN + col] + b_gate[3 * HIDN + col];
        const float wco = w_c[2 * HIDN + col];
        const float lw  = lin_w[col];
#pragma unroll
        for (int rt = 0; rt < 2; ++rt) {
#pragma unroll
            for (int r = 0; r < 8; ++r) {
                float I  = sigm(acc[rt][0][t][r] + bi);
                float Tg = tanhf(acc[rt][1][t][r] + bc);
                float Cc = I * Tg;
                float O  = sigm(acc[rt][2][t][r] + bo + wco * Cc);
                float Hh = O * tanhf(Cc);
                psum[rt][r] += Hh * lw;
            }
        }
    }
#pragma unroll
    for (int rt = 0; rt < 2; ++rt)
#pragma unroll
        for (int r = 0; r < 8; ++r)
            atomicAdd(&part[rt * 16 + half * 8 + r], psum[rt][r]);
    __syncthreads();

    if (tid < 32) out[node0 + tid] = sigm(part[tid] + lin_b[0]);
}

// ---------------- host launch ----------------

extern "C" void kernel_launch(void* const* d_in, const int* in_sizes, int n_in,
                              void* d_out, int out_size, void* d_ws, size_t ws_size,
                              hipStream_t stream) {
    const float*     x    = (const float*)d_in[0];
    const long long* ei   = (const long long*)d_in[1];   // int64 (2, E)
    const float*     Wx   = (const float*)d_in[2];       // (4,3,128,256)
    // d_in[3] = Wh: dead (H0 == 0)
    const float*     bx   = (const float*)d_in[4];
    const float*     bh   = (const float*)d_in[5];
    const float*     w_c  = (const float*)d_in[6];
    const float*     bg   = (const float*)d_in[7];
    const float*     lw   = (const float*)d_in[8];
    const float*     lb   = (const float*)d_in[9];
    float*           out  = (float*)d_out;

    // Workspace: deg/dis (N) | norm (E) | Tx1 (N*128) | Tx2 (N*128) | Pk (packed W)
    float*  deg  = (float*)d_ws;
    float*  norm = deg + NN;
    float*  Tx1  = norm + EE;
    float*  Tx2  = Tx1 + (size_t)NN * FIN;
    float2* Pk2  = (float2*)(Tx2 + (size_t)NN * FIN);
    size_t zero_bytes = (size_t)(NN + EE + 2 * (size_t)NN * FIN) * sizeof(float);

    hipMemsetAsync(d_ws, 0, zero_bytes, stream);

    pack_weights<<<(96 * 3 * 16 * 32 + 255) / 256, 256, 0, stream>>>(Wx, Pk2);

    deg_kernel <<<(EE + 255) / 256, 256, 0, stream>>>(ei, deg);
    dis_kernel <<<(NN + 255) / 256, 256, 0, stream>>>(deg);
    norm_kernel<<<(EE + 255) / 256, 256, 0, stream>>>(ei, deg, norm);

    const unsigned pthreads = (unsigned)EE * 32u;
    prop_kernel<<<(pthreads + 255) / 256, 256, 0, stream>>>(ei, norm, x, Tx1);    // Tx1 = L̂ x
    prop_kernel<<<(pthreads + 255) / 256, 256, 0, stream>>>(ei, norm, Tx1, Tx2);  // Tx2 = L̂ Tx1
    cheb2_finalize<<<(NN * FIN + 255) / 256, 256, 0, stream>>>(x, Tx2);           // Tx2 = 2*Tx2 - x

    fused_gemm_gates<<<NN / 32, 256, 0, stream>>>(x, Tx1, Tx2, Pk2,
                                                  bx, bh, w_c, bg, lw, lb, out);
}